// MoERouter_32933809225823
// MI455X (gfx1250) — compile-verified
//
#include <hip/hip_runtime.h>
#include <hip/hip_bf16.h>
#include <math.h>

typedef __attribute__((ext_vector_type(2))) float v2f;
typedef __attribute__((ext_vector_type(8))) float v8f;

namespace {
constexpr int HIDDEN        = 2048;
constexpr int NEXP          = 64;
constexpr int NTOK          = 16384;
constexpr int TOK_PER_BLOCK = 128;   // 8 waves * 16 tokens
constexpr int TOK_PER_WAVE  = 16;
constexpr int LDS_STRIDE    = NEXP + 1;   // pad to avoid bank conflicts

// d_out flat layout (reference tuple order)
constexpr size_t OFF_LOGITS = 0;
constexpr size_t OFF_W      = (size_t)NTOK * NEXP;            // 1048576
constexpr size_t OFF_SEL    = OFF_W + (size_t)NTOK * 2;       // 1081344
constexpr size_t OFF_MASK   = OFF_SEL + (size_t)NTOK * 2;     // 1114112
}

__global__ __launch_bounds__(256)
void moe_router_wmma(const float* __restrict__ X,
                     const float* __restrict__ Wg,
                     const float* __restrict__ bias,
                     float* __restrict__ out)
{
    __shared__ float lds_logits[TOK_PER_BLOCK * LDS_STRIDE];   // ~33 KB
    __shared__ int   lds_sel[TOK_PER_BLOCK * 2];

    const int lane = threadIdx.x & 31;
    const int wave = threadIdx.x >> 5;
    const int lo   = lane & 15;   // M (A) / N (B,C,D) within half-wave
    const int hi   = lane >> 4;   // selects K pair {0,1} vs {2,3}
    const int row0 = blockIdx.x * TOK_PER_BLOCK + wave * TOK_PER_WAVE;

    // ---------------- Phase 1: 16x64 logits tile via v_wmma_f32_16x16x4_f32 ----
    // A (16x4 f32): lane(0-15)=row M, regs K0,K1; lane(16-31)=row M, regs K2,K3
    // B (4x16 f32): mirrored -> each lane loads contiguous float2 from a W row
    const float* xp = X + (size_t)(row0 + lo) * HIDDEN + 2 * hi;
    const float* wp[4];
    #pragma unroll
    for (int nt = 0; nt < 4; ++nt)
        wp[nt] = Wg + (size_t)(nt * 16 + lo) * HIDDEN + 2 * hi;

    v8f acc[4] = {};

    #pragma unroll 4
    for (int k = 0; k < HIDDEN; k += 4) {
        const v2f a = *(const v2f*)(xp + k);
        #pragma unroll
        for (int nt = 0; nt < 4; ++nt) {
            const v2f bm = *(const v2f*)(wp[nt] + k);
            // 8 args: (neg_a, A, neg_b, B, c_mod, C, reuse_a, reuse_b)
            acc[nt] = __builtin_amdgcn_wmma_f32_16x16x4_f32(
                false, a, false, bm, (short)0, acc[nt], false, false);
        }
    }

    // Bias add + write logits to global and LDS.
    // C/D layout: VGPR r, lanes0-15 -> M=r, lanes16-31 -> M=r+8; N = lo.
    #pragma unroll
    for (int nt = 0; nt < 4; ++nt) {
        const float bv = bias[nt * 16 + lo];
        #pragma unroll
        for (int r = 0; r < 8; ++r) {
            const int   tl = wave * TOK_PER_WAVE + r + 8 * hi;  // token in block
            const float v  = acc[nt][r] + bv;
            lds_logits[tl * LDS_STRIDE + nt * 16 + lo] = v;
            out[OFF_LOGITS +
                (size_t)(blockIdx.x * TOK_PER_BLOCK + tl) * NEXP + nt * 16 + lo] = v;
        }
    }

    __syncthreads();

    // ---------------- Phase 2: top-2 + renormalized softmax weights ----------
    // renormalized top-k softmax == 2-way softmax over the top-2 logits.
    if (threadIdx.x < TOK_PER_BLOCK) {
        const int    t  = threadIdx.x;
        const float* lp = lds_logits + t * LDS_STRIDE;
        float b1 = -INFINITY, b2 = -INFINITY;
        int   i1 = 0, i2 = 0;
        #pragma unroll 4
        for (int e = 0; e < NEXP; ++e) {
            const float v = lp[e];
            if (v > b1)      { b2 = b1; i2 = i1; b1 = v; i1 = e; }
            else if (v > b2) { b2 = v;  i2 = e; }
        }
        const float e2  = expf(b2 - b1);
        const float inv = 1.0f / (1.0f + e2);
        const int   gt  = blockIdx.x * TOK_PER_BLOCK + t;
        out[OFF_W   + (size_t)gt * 2 + 0] = inv;
        out[OFF_W   + (size_t)gt * 2 + 1] = e2 * inv;
        out[OFF_SEL + (size_t)gt * 2 + 0] = (float)i1;
        out[OFF_SEL + (size_t)gt * 2 + 1] = (float)i2;
        lds_sel[t * 2 + 0] = i1;
        lds_sel[t * 2 + 1] = i2;
    }

    __syncthreads();

    // ---------------- Phase 3: one-hot expert mask [T, K, E], coalesced -------
    for (int i = threadIdx.x; i < TOK_PER_BLOCK * 2 * NEXP; i += 256) {
        const int t   = i >> 7;        // / (2*NEXP)
        const int rem = i & 127;
        const int kk  = rem >> 6;
        const int e   = rem & 63;
        const float v = (lds_sel[t * 2 + kk] == e) ? 1.0f : 0.0f;
        out[OFF_MASK +
            (size_t)(blockIdx.x * TOK_PER_BLOCK + t) * (2 * NEXP) + rem] = v;
    }
}

extern "C" void kernel_launch(void* const* d_in, const int* in_sizes, int n_in,
                              void* d_out, int out_size, void* d_ws, size_t ws_size,
                              hipStream_t stream) {
    const float* X  = (const float*)d_in[0];
    const float* Wg = (const float*)d_in[1];
    const float* b  = (const float*)d_in[2];
    float* out = (float*)d_out;

    dim3 grid(NTOK / TOK_PER_BLOCK);  // 128 blocks
    dim3 block(256);                  // 8 waves (wave32)
    moe_router_wmma<<<grid, block, 0, stream>>>(X, Wg, b, out);
}